// Quaternion_ResamplingModel_25013889532568
// MI455X (gfx1250) — compile-verified
//
#include <hip/hip_runtime.h>

// quatInterp / slerp resampling for MI455X (gfx1250, wave32).
//
// Shapes: quat (N=8192, B=256, 4) f32, ind (M=16384) f32, out (M,B,4) f32.
// Memory-bound (~100 MB HBM-effective @ 23.3 TB/s ~= 4-9 us) balanced against
// transcendental VALU; no matmul structure -> no WMMA.
//
// v2 changes vs v1:
//  - sinf/cosf -> __sinf/__cosf: argument is t*acos(cw) in [0, pi], so libm
//    range reduction is dead weight.  Native v_sin_f32/v_cos_f32 are TRANS32
//    ops that co-execute with the VALU FMA chain on CDNA5.
//  - 1/sqrtf -> rsqrtf (v_rsq_f32) for the two unit-norm normalizations and
//    the axis normalization.
// Kept: block = 256 lanes = one batch row, MS=8 m-values per block, uniform
// scalar index math (s_floor_f32/s_ceil_f32), coalesced global_load_b128,
// global_prefetch for next gather rows, NT global_store_b128.

typedef __attribute__((ext_vector_type(4))) float f4;

#define BQ 256   // batch dimension (fixed by reference: quat.shape[1] == 256)
#define MS 8     // m-values processed per block

__global__ __launch_bounds__(BQ) void quat_slerp_kernel(
    const f4* __restrict__ quat,    // [N*BQ] rows of float4 (w,x,y,z)
    const float* __restrict__ ind,  // [M]
    f4* __restrict__ out,           // [M*BQ]
    int N, int M)
{
    const int b = threadIdx.x;                    // 0..255, lane within row
    const int m_base = blockIdx.x * MS;
    const float nmax = (float)(N - 1);
    const float qnan = __builtin_nanf("");

#pragma unroll
    for (int i = 0; i < MS; ++i) {
        const int m = m_base + i;
        if (m >= M) return;

        // ---- uniform (per-block) index math: f, ind0, ind1, t -------------
        const float f = ind[m];                   // uniform -> scalar load
        int i0 = (int)floorf(f);
        int i1 = (int)ceilf(f);
        i0 = min(max(i0, 0), N - 1);
        i1 = min(max(i1, 0), N - 1);
        const float t = f - (float)i0;

        // ---- prefetch next iteration's gather rows (global_prefetch_b8) ---
        if (i + 1 < MS && m + 1 < M) {
            const float fn = ind[m + 1];
            int p0 = min(max((int)floorf(fn), 0), N - 1);
            int p1 = min(max((int)ceilf(fn),  0), N - 1);
            __builtin_prefetch((const void*)(quat + (size_t)p0 * BQ + b), 0, 3);
            __builtin_prefetch((const void*)(quat + (size_t)p1 * BQ + b), 0, 3);
        }

        // ---- coalesced 128-bit gathers ------------------------------------
        const f4 q0 = quat[(size_t)i0 * BQ + b];  // raw q0 (used in output)
        const f4 q1 = quat[(size_t)i1 * BQ + b];

        // ---- normalize both (inputs are ~unit norm; v_rsq_f32) -------------
        const float r0 = rsqrtf(q0.x*q0.x + q0.y*q0.y + q0.z*q0.z + q0.w*q0.w);
        const float r1 = rsqrtf(q1.x*q1.x + q1.y*q1.y + q1.z*q1.z + q1.w*q1.w);
        const float w1 = q0.x*r0, x1 = q0.y*r0, y1 = q0.z*r0, z1 = q0.w*r0;
        const float w2 = q1.x*r1, x2 = q1.y*r1, y2 = q1.z*r1, z2 = q1.w*r1;

        // ---- relative quaternion d ----------------------------------------
        float dw =  w1*w2 + x1*x2 + y1*y2 + z1*z2;
        float dx = -w1*x2 + x1*w2 - y1*z2 + z1*y2;
        float dy = -w1*y2 + x1*z2 + y1*w2 - z1*x2;
        float dz = -w1*z2 - x1*y2 + y1*x2 + z1*w2;
        if (dw < 0.0f) { dw = -dw; dx = -dx; dy = -dy; dz = -dz; }

        // ---- angle / axis --------------------------------------------------
        const float cw    = fminf(fmaxf(dw, -1.0f + 1e-6f), 1.0f - 1e-6f);
        const float half  = acosf(cw);            // angle = 2*acos; a/2 = t*acos
        const float angle = 2.0f * half;
        const bool  direct = angle < 1e-6f;

        const float axn2 = dx*dx + dy*dy + dz*dz;
        const float inv  = direct ? 1.0f : rsqrtf(axn2);   // 1/||axis||
        const float a2   = t * half;              // == (t*angle)/2, in [0, pi]
        const float s    = __sinf(a2);            // v_sin_f32 (TRANS32)
        const float c    = __cosf(a2);            // v_cos_f32 (TRANS32)
        const float qtw = c;
        const float qtx = dx * inv * s;
        const float qty = dy * inv * s;
        const float qtz = dz * inv * s;

        // ---- out = q0 (raw) * q_t0 ----------------------------------------
        f4 res;
        res.x = q0.x*qtw - q0.y*qtx - q0.z*qty - q0.w*qtz;
        res.y = q0.x*qtx + q0.y*qtw + q0.z*qtz - q0.w*qty;
        res.z = q0.x*qty - q0.y*qtz + q0.z*qtw + q0.w*qtx;
        res.w = q0.x*qtz + q0.y*qty - q0.z*qtx + q0.w*qtw;
        if (direct) res = q0;
        if (f < 0.0f || f > nmax) { res.x = qnan; res.y = qnan; res.z = qnan; res.w = qnan; }

        // ---- NT 128-bit store (write-once stream, keep L2 for quat) -------
        __builtin_nontemporal_store(res, out + (size_t)m * BQ + b);
    }
}

extern "C" void kernel_launch(void* const* d_in, const int* in_sizes, int n_in,
                              void* d_out, int out_size, void* d_ws, size_t ws_size,
                              hipStream_t stream) {
    (void)d_ws; (void)ws_size; (void)n_in; (void)out_size;
    const f4*    quat = (const f4*)d_in[0];
    const float* ind  = (const float*)d_in[1];
    f4*          out  = (f4*)d_out;

    const int M = in_sizes[1];                 // 16384
    const int N = in_sizes[0] / (BQ * 4);      // 8192

    const int grid = (M + MS - 1) / MS;
    quat_slerp_kernel<<<grid, BQ, 0, stream>>>(quat, ind, out, N, M);
}